// Model_5944234738327
// MI455X (gfx1250) — compile-verified
//
#include <hip/hip_runtime.h>

// ---------------------------------------------------------------------------
// Problem constants (match reference)
// ---------------------------------------------------------------------------
#define USER_N 100000
#define ITEM_N 200000
#define NN     300000            // USER_N + ITEM_N
#define FEATD  256
#define DIM    64
#define EDGES  3000000
#define EPSV   0.1f

typedef float v2f __attribute__((ext_vector_type(2)));
typedef float v8f __attribute__((ext_vector_type(8)));

// ---------------------------------------------------------------------------
// Small utility kernels
// ---------------------------------------------------------------------------
__global__ __launch_bounds__(256) void k_copy(float* __restrict__ dst,
                                              const float* __restrict__ src, int n) {
  int i = blockIdx.x * 256 + threadIdx.x;
  if (i < n) dst[i] = src[i];
}

__global__ __launch_bounds__(256) void k_zero(float* __restrict__ p, int n) {
  int i = blockIdx.x * 256 + threadIdx.x;
  if (i < n) p[i] = 0.f;
}

// v1 = v2 = ini (full N rows); main_user = ini[:USER]
__global__ __launch_bounds__(256) void k_init_out(const float* __restrict__ ini,
                                                  float* __restrict__ main_u,
                                                  float* __restrict__ v1,
                                                  float* __restrict__ v2, int n) {
  int i = blockIdx.x * 256 + threadIdx.x;
  if (i < n) {
    float v = ini[i];
    v1[i] = v;
    v2[i] = v;
    if (i < USER_N * DIM) main_u[i] = v;
  }
}

__global__ __launch_bounds__(256) void k_axpy(float* __restrict__ dst,
                                              const float* __restrict__ src, int n) {
  int i = blockIdx.x * 256 + threadIdx.x;
  if (i < n) dst[i] += src[i];
}

// ---------------------------------------------------------------------------
// Fused item MLP + row L2-norm, using V_WMMA_F32_16X16X4_F32.
//   H = relu(X @ W1 + b1)   [16x64 tile per wave, K=256]
//   Y = H @ W2 + b2         [via LDS transpose of H, K=64]
//   item_n = l2norm(Y)      -> written to d_out slot 1 and ini[USER:]
// Wave32: A operand = 16x4 fp32 (2 VGPRs/lane), C/D = 16x16 fp32 (8 VGPRs).
// ---------------------------------------------------------------------------
__global__ __launch_bounds__(256)
void k_mlp_item(const float* __restrict__ X,    // [ITEM, 256]
                const float* __restrict__ W1,   // [256, 64]
                const float* __restrict__ b1,   // [64]
                const float* __restrict__ W2,   // [64, 64]
                const float* __restrict__ b2,   // [64]
                float* __restrict__ item_n_out, // [ITEM, 64]
                float* __restrict__ ini)        // [N, 64] (write rows USER..N-1)
{
  __shared__ float w2s[DIM * DIM];      // 16 KB
  __shared__ float hs[8][16 * 68];      // 34 KB : per-wave 16x64 H tile (padded)

  for (int i = threadIdx.x; i < DIM * DIM; i += 256) w2s[i] = W2[i];
  __syncthreads();                      // no further block-wide syncs below

  const int wave = threadIdx.x >> 5;
  const int lane = threadIdx.x & 31;
  const int hf   = lane >> 4;           // which 16-lane half
  const int l16  = lane & 15;

  const int tile = blockIdx.x * 8 + wave;     // 16-row output tile
  if (tile >= ITEM_N / 16) return;
  const int row0 = tile * 16;

  // ---- GEMM1: acc = X_tile @ W1 ------------------------------------------
  v8f acc[4] = {v8f{}, v8f{}, v8f{}, v8f{}};
  const float* xrow = X + (size_t)(row0 + l16) * FEATD;   // A row for this lane
  for (int k = 0; k < FEATD; k += 4) {
    const int ka = k + hf * 2;                            // K pair per half
    v2f a;
    a.x = xrow[ka];
    a.y = xrow[ka + 1];
#pragma unroll
    for (int t = 0; t < 4; ++t) {
      v2f b;
      b.x = W1[(size_t)ka       * DIM + t * 16 + l16];
      b.y = W1[(size_t)(ka + 1) * DIM + t * 16 + l16];
      acc[t] = __builtin_amdgcn_wmma_f32_16x16x4_f32(false, a, false, b,
                                                     (short)0, acc[t], false, false);
    }
  }

  // ---- bias + relu, stage H in LDS in row-major (A layout for GEMM2) -----
  float* H = &hs[wave][0];
#pragma unroll
  for (int t = 0; t < 4; ++t) {
    const int c = t * 16 + l16;
    const float bias = b1[c];
#pragma unroll
    for (int i = 0; i < 8; ++i) {
      const int r = i + hf * 8;
      float v = acc[t][i] + bias;
      H[r * 68 + c] = v > 0.f ? v : 0.f;
    }
  }

  // ---- GEMM2: acc2 = H @ W2 (K = 64, W2 from LDS) ------------------------
  v8f acc2[4] = {v8f{}, v8f{}, v8f{}, v8f{}};
  for (int k = 0; k < DIM; k += 4) {
    const int ka = k + hf * 2;
    v2f a;
    a.x = H[l16 * 68 + ka];
    a.y = H[l16 * 68 + ka + 1];
#pragma unroll
    for (int t = 0; t < 4; ++t) {
      v2f b;
      b.x = w2s[ka       * DIM + t * 16 + l16];
      b.y = w2s[(ka + 1) * DIM + t * 16 + l16];
      acc2[t] = __builtin_amdgcn_wmma_f32_16x16x4_f32(false, a, false, b,
                                                      (short)0, acc2[t], false, false);
    }
  }

  // ---- bias b2 -----------------------------------------------------------
#pragma unroll
  for (int t = 0; t < 4; ++t) {
    const float bias = b2[t * 16 + l16];
#pragma unroll
    for (int i = 0; i < 8; ++i) acc2[t][i] += bias;
  }

  // ---- row L2-norm + stores ---------------------------------------------
  // Row (i + hf*8) lives in VGPR i across the 16 lanes of this half.
#pragma unroll
  for (int i = 0; i < 8; ++i) {
    float ss = 0.f;
#pragma unroll
    for (int t = 0; t < 4; ++t) ss += acc2[t][i] * acc2[t][i];
    ss += __shfl_xor(ss, 1, 16);
    ss += __shfl_xor(ss, 2, 16);
    ss += __shfl_xor(ss, 4, 16);
    ss += __shfl_xor(ss, 8, 16);
    const float inv = 1.f / fmaxf(sqrtf(ss), 1e-12f);
    const int rg = row0 + i + hf * 8;
#pragma unroll
    for (int t = 0; t < 4; ++t) {
      const float v = acc2[t][i] * inv;
      const int c = t * 16 + l16;
      item_n_out[(size_t)rg * DIM + c]          = v;
      ini[(size_t)(USER_N + rg) * DIM + c]      = v;
    }
  }
}

// ---------------------------------------------------------------------------
// SpMM: out[r,:] += vals[e] * x[cols[e],:].  16 lanes per edge, float4 per
// lane, scatter via global_atomic_add_f32 (executes at L2; x fits in 192MB L2).
// ---------------------------------------------------------------------------
__global__ __launch_bounds__(256)
void k_spmm(const int* __restrict__ rows, const int* __restrict__ cols,
            const float* __restrict__ vals, const float* __restrict__ x,
            float* __restrict__ out) {
  const int t = blockIdx.x * 256 + threadIdx.x;
  const int e = t >> 4;
  if (e >= EDGES) return;
  const int sub = (t & 15) * 4;
  const int r = rows[e];
  const int c = cols[e];
  const float v = vals[e];
  const float4 xv = *(const float4*)(x + (size_t)c * DIM + sub);
  float* o = out + (size_t)r * DIM + sub;
  atomicAdd(o + 0, v * xv.x);
  atomicAdd(o + 1, v * xv.y);
  atomicAdd(o + 2, v * xv.z);
  atomicAdd(o + 3, v * xv.w);
}

// ---------------------------------------------------------------------------
// Fused: e += l2norm(noise_row) * sign(e) * EPS ;  acc += e
// One wave32 per row; 2 dims per lane; shuffle reduction for the row norm.
// ---------------------------------------------------------------------------
__global__ __launch_bounds__(256)
void k_perturb_acc(float* __restrict__ e, const float* __restrict__ noise,
                   float* __restrict__ acc) {
  const int row  = (blockIdx.x * 256 + threadIdx.x) >> 5;
  const int lane = threadIdx.x & 31;
  if (row >= NN) return;
  const size_t base = (size_t)row * DIM;

  const float n0 = noise[base + lane];
  const float n1 = noise[base + lane + 32];
  float ss = n0 * n0 + n1 * n1;
  ss += __shfl_xor(ss, 1, 32);
  ss += __shfl_xor(ss, 2, 32);
  ss += __shfl_xor(ss, 4, 32);
  ss += __shfl_xor(ss, 8, 32);
  ss += __shfl_xor(ss, 16, 32);
  const float inv = 1.f / fmaxf(sqrtf(ss), 1e-12f);

  float v0 = e[base + lane];
  float s0 = (v0 > 0.f) ? 1.f : ((v0 < 0.f) ? -1.f : 0.f);
  v0 += n0 * inv * s0 * EPSV;
  e[base + lane] = v0;
  acc[base + lane] += v0;

  float v1 = e[base + lane + 32];
  float s1 = (v1 > 0.f) ? 1.f : ((v1 < 0.f) ? -1.f : 0.f);
  v1 += n1 * inv * s1 * EPSV;
  e[base + lane + 32] = v1;
  acc[base + lane + 32] += v1;
}

// ---------------------------------------------------------------------------
// Orchestration
// ---------------------------------------------------------------------------
extern "C" void kernel_launch(void* const* d_in, const int* in_sizes, int n_in,
                              void* d_out, int out_size, void* d_ws, size_t ws_size,
                              hipStream_t stream) {
  const float* uEmbeds = (const float*)d_in[0];
  const float* X       = (const float*)d_in[1];
  const float* W1      = (const float*)d_in[2];
  const float* b1      = (const float*)d_in[3];
  const float* W2      = (const float*)d_in[4];
  const float* b2      = (const float*)d_in[5];
  const float* vals    = (const float*)d_in[6];
  const float* noise1  = (const float*)d_in[7];
  const float* noise2  = (const float*)d_in[8];
  const int*   rows    = (const int*)d_in[9];
  const int*   cols    = (const int*)d_in[10];

  float* out    = (float*)d_out;
  float* main_u = out;                                   // [USER,64]
  float* item_n = out + (size_t)USER_N * DIM;            // [ITEM,64]
  float* v1     = out + (size_t)NN * DIM;                // [N,64] (user‖item)
  float* v2     = out + 2 * (size_t)NN * DIM;            // [N,64]

  float* ws   = (float*)d_ws;
  float* ini  = ws;                                      // [N,64]
  float* bufA = ws + (size_t)NN * DIM;                   // [N,64]
  float* bufB = ws + 2 * (size_t)NN * DIM;               // [N,64]

  const int NV = NN * DIM;                               // 19.2M
  const int gNV   = (NV + 255) / 256;                    // 75000
  const int gUser = (USER_N * DIM + 255) / 256;          // 25000
  const int gSpmm = (EDGES * 16 + 255) / 256;            // 187500
  const int gRow  = (NN + 7) / 8;                        // 37500 (wave/row)
  const int gMlp  = (ITEM_N / 16 + 7) / 8;               // 1563

  // ini = concat(uEmbeds, l2norm(mlp(item_feats)))
  k_copy<<<gUser, 256, 0, stream>>>(ini, uEmbeds, USER_N * DIM);
  k_mlp_item<<<gMlp, 256, 0, stream>>>(X, W1, b1, W2, b2, item_n, ini);
  k_init_out<<<gNV, 256, 0, stream>>>(ini, main_u, v1, v2, NV);

  // ---- main view: acc_user += spmm^l(ini) --------------------------------
  k_zero<<<gNV, 256, 0, stream>>>(bufA, NV);
  k_spmm<<<gSpmm, 256, 0, stream>>>(rows, cols, vals, ini, bufA);
  k_axpy<<<gUser, 256, 0, stream>>>(main_u, bufA, USER_N * DIM);

  k_zero<<<gNV, 256, 0, stream>>>(bufB, NV);
  k_spmm<<<gSpmm, 256, 0, stream>>>(rows, cols, vals, bufA, bufB);
  k_axpy<<<gUser, 256, 0, stream>>>(main_u, bufB, USER_N * DIM);

  k_zero<<<gNV, 256, 0, stream>>>(bufA, NV);
  k_spmm<<<gSpmm, 256, 0, stream>>>(rows, cols, vals, bufB, bufA);
  k_axpy<<<gUser, 256, 0, stream>>>(main_u, bufA, USER_N * DIM);

  // ---- perturbed view 1 --------------------------------------------------
  k_zero<<<gNV, 256, 0, stream>>>(bufA, NV);
  k_spmm<<<gSpmm, 256, 0, stream>>>(rows, cols, vals, ini, bufA);
  k_perturb_acc<<<gRow, 256, 0, stream>>>(bufA, noise1 + 0 * (size_t)NV, v1);

  k_zero<<<gNV, 256, 0, stream>>>(bufB, NV);
  k_spmm<<<gSpmm, 256, 0, stream>>>(rows, cols, vals, bufA, bufB);
  k_perturb_acc<<<gRow, 256, 0, stream>>>(bufB, noise1 + 1 * (size_t)NV, v1);

  k_zero<<<gNV, 256, 0, stream>>>(bufA, NV);
  k_spmm<<<gSpmm, 256, 0, stream>>>(rows, cols, vals, bufB, bufA);
  k_perturb_acc<<<gRow, 256, 0, stream>>>(bufA, noise1 + 2 * (size_t)NV, v1);

  // ---- perturbed view 2 --------------------------------------------------
  k_zero<<<gNV, 256, 0, stream>>>(bufA, NV);
  k_spmm<<<gSpmm, 256, 0, stream>>>(rows, cols, vals, ini, bufA);
  k_perturb_acc<<<gRow, 256, 0, stream>>>(bufA, noise2 + 0 * (size_t)NV, v2);

  k_zero<<<gNV, 256, 0, stream>>>(bufB, NV);
  k_spmm<<<gSpmm, 256, 0, stream>>>(rows, cols, vals, bufA, bufB);
  k_perturb_acc<<<gRow, 256, 0, stream>>>(bufB, noise2 + 1 * (size_t)NV, v2);

  k_zero<<<gNV, 256, 0, stream>>>(bufA, NV);
  k_spmm<<<gSpmm, 256, 0, stream>>>(rows, cols, vals, bufB, bufA);
  k_perturb_acc<<<gRow, 256, 0, stream>>>(bufA, noise2 + 2 * (size_t)NV, v2);

  (void)in_sizes; (void)n_in; (void)out_size; (void)ws_size;
}